// contextual_layers_16569983828440
// MI455X (gfx1250) — compile-verified
//
#include <hip/hip_runtime.h>
#include <hip/hip_bf16.h>
#include <climits>

// ---------------------------------------------------------------------------
// 3-layer GAT on MI455X (gfx1250, wave32).
// GEMMs use V_WMMA_F32_16X16X4_F32 (full f32 precision; GEMM is ~3 GFLOP total
// and NOT the bottleneck -- the edge scatter/gather phase is, and it is
// L2-resident on the 192MB L2). Everything stays f32 end-to-end.
// ---------------------------------------------------------------------------

#define N_NODES 100000
#define N_EDGES 1600000
#define IN_FEATS 128
#define H_FEATS 64

typedef __attribute__((ext_vector_type(2))) float v2f;
typedef __attribute__((ext_vector_type(8))) float v8f;

// ---------------- ordered-int float max trick (exact float ordering) --------
__device__ __forceinline__ int f2ord(float f) {
  int i = __float_as_int(f);
  return i >= 0 ? i : (i ^ 0x7fffffff);
}
__device__ __forceinline__ float ord2f(int i) {
  return __int_as_float(i >= 0 ? i : (i ^ 0x7fffffff));
}

// ---------------- WMMA f32 GEMM: H[nRows,64] = X[nRows,K] @ W[K,64] ---------
// One wave computes one 16x16 tile of H. Block = 256 thr = 8 waves
// = 2 M-tiles x 4 N-tiles  -> block covers 32 rows x 64 cols.
// A (16x4 f32):  lane l: row = l&15, holds K = (l>>4)*2 .. +1   (ISA 7.12.2)
// B (4x16 f32):  lane l: col = l&15, holds K = (l>>4)*2 .. +1
// C/D (16x16):   VGPR v: row = v + 8*(l>>4), col = l&15
__global__ __launch_bounds__(256) void gat_gemm_wmma_f32(
    const float* __restrict__ X, const float* __restrict__ W,
    float* __restrict__ H, int nRows, int K) {
  const int lane = threadIdx.x & 31;
  const int wave = threadIdx.x >> 5;
  const int mTile = blockIdx.x * 2 + (wave >> 2);
  const int nTile = wave & 3;
  const int row0 = mTile * 16;
  const int col0 = nTile * 16;
  if (row0 >= nRows) return;  // wave-uniform: EXEC stays all-ones for WMMA

  const int r = lane & 15;            // A-row / B-col within tile
  const int khalf = (lane >> 4) * 2;  // k sub-offset {0,2}

  v8f acc = {};
  const float* xrow = X + (size_t)(row0 + r) * K;
  const float* wcol = W + (size_t)(col0 + r);

#pragma unroll 4
  for (int k = 0; k < K; k += 4) {
    v2f a = *(const v2f*)(xrow + k + khalf);  // 8B-aligned (khalf even, K%4==0)
    v2f b;
    b.x = wcol[(size_t)(k + khalf) * H_FEATS];
    b.y = wcol[(size_t)(k + khalf + 1) * H_FEATS];
    // 8 args: (neg_a, A, neg_b, B, c_mod, C, reuse_a, reuse_b)
    acc = __builtin_amdgcn_wmma_f32_16x16x4_f32(
        false, a, false, b, (short)0, acc, false, false);
  }

  const int mOff = (lane >> 4) * 8;
  const int n = lane & 15;
  float* hp = H + (size_t)(row0 + mOff) * H_FEATS + col0 + n;
#pragma unroll
  for (int v = 0; v < 8; ++v) hp[(size_t)v * H_FEATS] = acc[v];
}

// ---------------- per-node attention scores (wave per node, wave32) ---------
__global__ __launch_bounds__(256) void gat_scores(
    const float* __restrict__ H, const float* __restrict__ al,
    const float* __restrict__ ar, float* __restrict__ el,
    float* __restrict__ er, int nNodes) {
  const int gw = (int)((blockIdx.x * blockDim.x + threadIdx.x) >> 5);
  const int lane = threadIdx.x & 31;
  if (gw >= nNodes) return;
  const float2 h = *(const float2*)(H + (size_t)gw * H_FEATS + lane * 2);
  const float2 l2 = *(const float2*)(al + lane * 2);
  const float2 r2 = *(const float2*)(ar + lane * 2);
  float pl = h.x * l2.x + h.y * l2.y;
  float pr = h.x * r2.x + h.y * r2.y;
#pragma unroll
  for (int off = 16; off > 0; off >>= 1) {
    pl += __shfl_down(pl, off, 32);
    pr += __shfl_down(pr, off, 32);
  }
  if (lane == 0) { el[gw] = pl; er[gw] = pr; }
}

// ---------------- per-layer node state init --------------------------------
__global__ __launch_bounds__(256) void gat_init_nodes(
    int* __restrict__ m, float* __restrict__ s, int nNodes) {
  const int i = blockIdx.x * blockDim.x + threadIdx.x;
  if (i < nNodes) { m[i] = INT_MIN; s[i] = 0.0f; }
}

__global__ __launch_bounds__(256) void gat_zero_f32(float* __restrict__ p, int n) {
  const int i = blockIdx.x * blockDim.x + threadIdx.x;
  if (i < n) p[i] = 0.0f;
}

// ---------------- edge pass 1: leaky-relu score + segment max ---------------
__global__ __launch_bounds__(256) void gat_edge_max(
    const int* __restrict__ src, const int* __restrict__ dst,
    const float* __restrict__ el, const float* __restrict__ er,
    float* __restrict__ ebuf, int* __restrict__ m, int nEdges) {
  const int i = blockIdx.x * blockDim.x + threadIdx.x;
  if (i >= nEdges) return;
  const int sN = src[i], dN = dst[i];
  float e = el[sN] + er[dN];
  e = e > 0.0f ? e : 0.2f * e;  // leaky_relu, NEG_SLOPE = 0.2
  ebuf[i] = e;
  atomicMax(m + dN, f2ord(e));
}

// ---------------- edge pass 2: exp(e - m[dst]) + segment sum ----------------
__global__ __launch_bounds__(256) void gat_edge_expsum(
    const int* __restrict__ dst, float* __restrict__ ebuf,
    const int* __restrict__ m, float* __restrict__ s, int nEdges) {
  const int i = blockIdx.x * blockDim.x + threadIdx.x;
  if (i >= nEdges) return;
  const int dN = dst[i];
  const float ex = __expf(ebuf[i] - ord2f(m[dN]));
  ebuf[i] = ex;
  atomicAdd(s + dN, ex);
}

// ---------------- edge pass 3: out[dst] += alpha * h[src] (wave per edge) ---
__global__ __launch_bounds__(256) void gat_edge_aggr(
    const int* __restrict__ src, const int* __restrict__ dst,
    const float* __restrict__ ebuf, const float* __restrict__ ssum,
    const float* __restrict__ H, float* __restrict__ accum, int nEdges) {
  const int gw = (int)((blockIdx.x * blockDim.x + threadIdx.x) >> 5);
  const int lane = threadIdx.x & 31;
  if (gw >= nEdges) return;
  const int sN = src[gw], dN = dst[gw];
  const float alpha = ebuf[gw] / ssum[dN];
  const float2 hv = *(const float2*)(H + (size_t)sN * H_FEATS + lane * 2);
  float* op = accum + (size_t)dN * H_FEATS + lane * 2;
  atomicAdd(op + 0, alpha * hv.x);
  atomicAdd(op + 1, alpha * hv.y);
}

// ---------------- finalize: relu(accum + b) ---------------------------------
__global__ __launch_bounds__(256) void gat_finalize(
    const float* __restrict__ accum, const float* __restrict__ b,
    float* __restrict__ out, int nElems) {
  const int i = blockIdx.x * blockDim.x + threadIdx.x;
  if (i >= nElems) return;
  const float v = accum[i] + b[i & (H_FEATS - 1)];
  out[i] = v > 0.0f ? v : 0.0f;
}

// ---------------------------------------------------------------------------
static void run_layer(const float* X, int K, const float* W, const float* al,
                      const float* ar, const float* b, const int* src,
                      const int* dst, float* H, float* accum, float* ebuf,
                      float* el, float* er, float* ssum, int* mbuf, float* out,
                      hipStream_t stream) {
  const int N = N_NODES, E = N_EDGES;
  // GEMM first: accum may alias the GEMM input (stream order protects us).
  gat_gemm_wmma_f32<<<N / 32, 256, 0, stream>>>(X, W, H, N, K);
  gat_init_nodes<<<(N + 255) / 256, 256, 0, stream>>>(mbuf, ssum, N);
  gat_zero_f32<<<(N * H_FEATS + 255) / 256, 256, 0, stream>>>(accum, N * H_FEATS);
  gat_scores<<<(N * 32 + 255) / 256, 256, 0, stream>>>(H, al, ar, el, er, N);
  gat_edge_max<<<(E + 255) / 256, 256, 0, stream>>>(src, dst, el, er, ebuf, mbuf, E);
  gat_edge_expsum<<<(E + 255) / 256, 256, 0, stream>>>(dst, ebuf, mbuf, ssum, E);
  gat_edge_aggr<<<(E * 32 + 255) / 256, 256, 0, stream>>>(src, dst, ebuf, ssum, H, accum, E);
  gat_finalize<<<(N * H_FEATS + 255) / 256, 256, 0, stream>>>(accum, b, out, N * H_FEATS);
}

extern "C" void kernel_launch(void* const* d_in, const int* in_sizes, int n_in,
                              void* d_out, int out_size, void* d_ws, size_t ws_size,
                              hipStream_t stream) {
  const float* in_feat = (const float*)d_in[0];
  const int* src = (const int*)d_in[1];
  const int* dst = (const int*)d_in[2];
  const float* W1 = (const float*)d_in[3];
  const float* al1 = (const float*)d_in[4];
  const float* ar1 = (const float*)d_in[5];
  const float* b1 = (const float*)d_in[6];
  const float* W2 = (const float*)d_in[7];
  const float* al2 = (const float*)d_in[8];
  const float* ar2 = (const float*)d_in[9];
  const float* b2 = (const float*)d_in[10];
  const float* W3 = (const float*)d_in[11];
  const float* al3 = (const float*)d_in[12];
  const float* ar3 = (const float*)d_in[13];
  const float* b3 = (const float*)d_in[14];

  // Workspace carve-up (~59 MB total).
  float* P0 = (float*)d_ws;                       // [N,64] GEMM output h
  float* P1 = P0 + (size_t)N_NODES * H_FEATS;     // [N,64] accum / next input
  float* ebuf = P1 + (size_t)N_NODES * H_FEATS;   // [E] e then ex
  float* el = ebuf + (size_t)N_EDGES;             // [N]
  float* er = el + N_NODES;                       // [N]
  float* ssum = er + N_NODES;                     // [N]
  int* mbuf = (int*)(ssum + N_NODES);             // [N]

  float* out = (float*)d_out;

  // Layer 1: h1 = relu(aggr(in_feat @ W1) + b1) -> P1
  run_layer(in_feat, IN_FEATS, W1, al1, ar1, b1, src, dst,
            P0, P1, ebuf, el, er, ssum, mbuf, P1, stream);
  // Layer 2: h2 = relu(aggr(P1 @ W2) + b2) -> P1 (in place after GEMM to P0)
  run_layer(P1, H_FEATS, W2, al2, ar2, b2, src, dst,
            P0, P1, ebuf, el, er, ssum, mbuf, P1, stream);
  // Layer 3: out = relu(aggr(P1 @ W3) + b3) -> d_out
  run_layer(P1, H_FEATS, W3, al3, ar3, b3, src, dst,
            P0, out, ebuf, el, er, ssum, mbuf, out, stream);
}